// CustomModel_7378753814834
// MI455X (gfx1250) — compile-verified
//
#include <hip/hip_runtime.h>

typedef __attribute__((ext_vector_type(2))) float v2f;
typedef __attribute__((ext_vector_type(4))) float v4f;
typedef __attribute__((ext_vector_type(8))) float v8f;

#define B_ 128
#define K_ 49
#define D_ 2048

// sigma = 1..5 -> 1/(2 sigma^2)
#define INV0 0.5f
#define INV1 0.125f
#define INV2 (1.0f / 18.0f)
#define INV3 0.03125f
#define INV4 0.02f

// ---------------------------------------------------------------------------
// Phase 1: d2[b,d] = sum_k (x[b,k,d]-w[k,d])^2 via V_WMMA_F32_16X16X4_F32.
// One wave32 handles 16 consecutive d columns of one batch row b.
// A-matrix (16x4 f32): lane<16 -> {K=k0, K=k0+1}, lane>=16 -> {K=k0+2, K=k0+3},
// row M = lane&15 = d offset. B = ones(4x16) so D = A*ones + C accumulates the
// K-sum into every column of C. Result for M=m lives in VGPR (m&7) of lanes
// 0-15 (m<8) / 16-31 (m>=8); lanes 0 and 16 extract and apply exp per sigma.
// ---------------------------------------------------------------------------
__global__ __launch_bounds__(256) void d2_kv_wmma_kernel(
    const float* __restrict__ x, const float* __restrict__ w,
    float* __restrict__ dst, int kvMode)
{
    const int tid    = threadIdx.x;
    const int lane   = tid & 31;
    const int waveId = blockIdx.x * 8 + (tid >> 5);   // 8 waves / block
    const int b      = waveId >> 7;                   // 128 d-tiles per row
    const int dtile  = waveId & 127;
    const int d0     = dtile << 4;

    const int half = lane >> 4;
    const int m    = lane & 15;
    const int d    = d0 + m;

    const float* __restrict__ xb = x + (size_t)b * (K_ * D_);

    v8f c = {};                       // f32 16x16 accumulator (8 VGPRs)
    v2f ones; ones.x = 1.0f; ones.y = 1.0f;

    // chunks 0..11 cover k = 0..47, always in range
#pragma unroll
    for (int chunk = 0; chunk < 12; ++chunk) {
        const int k0 = chunk * 4 + half * 2;
        float dx0 = xb[(size_t)k0 * D_ + d]       - w[(size_t)k0 * D_ + d];
        float dx1 = xb[(size_t)(k0 + 1) * D_ + d] - w[(size_t)(k0 + 1) * D_ + d];
        v2f a; a.x = dx0 * dx0; a.y = dx1 * dx1;
        c = __builtin_amdgcn_wmma_f32_16x16x4_f32(false, a, false, ones,
                                                  (short)0, c, false, false);
    }
    // tail chunk: k = 48..51, only k = 48 (half==0, slot 0) is valid
    {
        float s0 = 0.0f;
        if (half == 0) {
            float dx = xb[(size_t)48 * D_ + d] - w[(size_t)48 * D_ + d];
            s0 = dx * dx;
        }
        v2f a; a.x = s0; a.y = 0.0f;
        c = __builtin_amdgcn_wmma_f32_16x16x4_f32(false, a, false, ones,
                                                  (short)0, c, false, false);
    }

    // Extract: lane 0 holds M=0..7 in c[0..7]; lane 16 holds M=8..15.
    if ((lane & 15) == 0) {
        const int dbase = d0 + half * 8;
        if (kvMode) {
#pragma unroll
            for (int j = 0; j < 8; ++j) {
                const float d2v = c[j];
                const int   dd  = dbase + j;
                dst[((size_t)0 * B_ + b) * D_ + dd] = expf(-d2v * INV0);
                dst[((size_t)1 * B_ + b) * D_ + dd] = expf(-d2v * INV1);
                dst[((size_t)2 * B_ + b) * D_ + dd] = expf(-d2v * INV2);
                dst[((size_t)3 * B_ + b) * D_ + dd] = expf(-d2v * INV3);
                dst[((size_t)4 * B_ + b) * D_ + dd] = expf(-d2v * INV4);
            }
        } else {
#pragma unroll
            for (int j = 0; j < 8; ++j)
                dst[(size_t)b * D_ + dbase + j] = c[j];
        }
    }
}

// ---------------------------------------------------------------------------
// Phase 2: broadcast kv[s,b,:] into 49 K-rows of the output.
// Grid (2, 49, 640): z = s*128+b, y = k, x*256+tid = float4 index along D.
// kv (5 MB) stays L2-resident across the 49x reuse; output stores are
// nontemporal (never re-read, 257 MB >> L2).
// ---------------------------------------------------------------------------
__global__ __launch_bounds__(256) void bcast_kernel(
    const float* __restrict__ src, float* __restrict__ out, int kvMode)
{
    const int d4 = blockIdx.x * 256 + threadIdx.x;    // 0..511
    const int k  = blockIdx.y;                        // 0..48
    const int sb = blockIdx.z;                        // 0..639 (s*128+b)

    const v4f* __restrict__ src4 = (const v4f*)src;
    v4f v;
    if (kvMode) {
        v = src4[(size_t)sb * (D_ / 4) + d4];
    } else {
        const int   b = sb & (B_ - 1);
        const int   s = sb >> 7;
        const float invs[5] = {INV0, INV1, INV2, INV3, INV4};
        const float inv = invs[s];
        v4f t = src4[(size_t)b * (D_ / 4) + d4];
        v.x = expf(-t.x * inv);
        v.y = expf(-t.y * inv);
        v.z = expf(-t.z * inv);
        v.w = expf(-t.w * inv);
    }
    v4f* dst4 = (v4f*)out + ((size_t)sb * K_ + k) * (D_ / 4) + d4;
    __builtin_nontemporal_store(v, dst4);
}

extern "C" void kernel_launch(void* const* d_in, const int* in_sizes, int n_in,
                              void* d_out, int out_size, void* d_ws, size_t ws_size,
                              hipStream_t stream) {
    (void)in_sizes; (void)n_in; (void)out_size;
    const float* x = (const float*)d_in[0];   // (128, 49, 2048) f32
    const float* w = (const float*)d_in[1];   // (1, 49, 2048) f32
    float*       out = (float*)d_out;         // (640, 49, 2048) f32
    float*       ws  = (float*)d_ws;

    const size_t needKv = (size_t)5 * B_ * D_ * sizeof(float);   // 5.25 MB
    const int kvMode = (ws_size >= needKv) ? 1 : 0;              // else raw d2 (1 MB)

    // Phase 1: 128 rows * 128 d-tiles = 16384 waves = 2048 blocks of 8 waves
    d2_kv_wmma_kernel<<<2048, 256, 0, stream>>>(x, w, ws, kvMode);

    // Phase 2: 2 * 49 * 640 blocks, each stores 4 KB contiguously
    dim3 g2(2, K_, 5 * B_);
    bcast_kernel<<<g2, 256, 0, stream>>>(ws, out, kvMode);
}